// SimpleGCN_57372173140576
// MI455X (gfx1250) — compile-verified
//
#include <hip/hip_runtime.h>
#include <hip/hip_bf16.h>

typedef __attribute__((ext_vector_type(2))) float v2f;
typedef __attribute__((ext_vector_type(8))) float v8f;

#define DIN   128
#define DH    128
#define DOUT  40
#define NGR   512

static inline long long cdivll(long long a, long long b) { return (a + b - 1) / b; }

// ---------------------------------------------------------------- fill
__global__ void fill_kernel(float* __restrict__ p, float v, long long n) {
    long long id = (long long)blockIdx.x * blockDim.x + threadIdx.x;
    if (id < n) p[id] = v;
}

// ---------------------------------------------------------------- degree
__global__ void deg_kernel(const int* __restrict__ dst, float* __restrict__ deg, int E) {
    int e = blockIdx.x * blockDim.x + threadIdx.x;
    if (e < E) atomicAdd(&deg[dst[e]], 1.0f);
}

__global__ void dinv_kernel(float* __restrict__ deg, int n) {
    int i = blockIdx.x * blockDim.x + threadIdx.x;
    if (i < n) deg[i] = rsqrtf(deg[i]);   // deg >= 1 always
}

// ---------------------------------------------------------------- GEMM1: Y[N,128] = X[N,128] @ W[128,128]
// block: 256 threads = 8 waves; block handles 16 rows; wave w handles cols [16w,16w+16)
__global__ __launch_bounds__(256) void gemm1_kernel(const float* __restrict__ X,
                                                    const float* __restrict__ W,
                                                    float* __restrict__ Y) {
    __shared__ float xs[16][DIN + 4];
    const int row0 = blockIdx.x * 16;
    const int tid  = threadIdx.x;

    // cooperative load of 16x128 tile (512 float4 across 256 threads)
    for (int i = tid; i < 512; i += 256) {
        int r  = i >> 5;            // 32 float4 per row
        int c4 = (i & 31) << 2;
        float4 v = *(const float4*)(X + (size_t)(row0 + r) * DIN + c4);
        *(float4*)&xs[r][c4] = v;
    }
    __syncthreads();

    const int wave  = tid >> 5;
    const int lane  = tid & 31;
    const int n0    = wave * 16;
    const int m     = lane & 15;            // A row / B-C-D column within tile
    const int khalf = (lane >> 4) << 1;     // lanes 16..31 carry K+2,K+3

    v8f c = {};
    for (int k0 = 0; k0 < DIN; k0 += 4) {
        v2f a, b;
        a.x = xs[m][k0 + khalf];
        a.y = xs[m][k0 + khalf + 1];
        b.x = W[(size_t)(k0 + khalf)     * DH + n0 + m];
        b.y = W[(size_t)(k0 + khalf + 1) * DH + n0 + m];
        c = __builtin_amdgcn_wmma_f32_16x16x4_f32(false, a, false, b, (short)0, c, false, false);
    }

    const int rbase = row0 + ((lane >> 4) << 3);   // upper lane half holds rows M=8..15
#pragma unroll
    for (int i = 0; i < 8; ++i)
        Y[(size_t)(rbase + i) * DH + n0 + m] = c[i];
}

// ---------------------------------------------------------------- GEMM2: Y[N,40] = H[N,128] @ W[128,40]
// block: 128 threads = 4 waves; waves 0..2 cover the 3 column tiles (cols 0..47, guarded at 40)
__global__ __launch_bounds__(128) void gemm2_kernel(const float* __restrict__ H,
                                                    const float* __restrict__ W,
                                                    float* __restrict__ Y) {
    __shared__ float hs[16][DH + 4];
    const int row0 = blockIdx.x * 16;
    const int tid  = threadIdx.x;

    for (int i = tid; i < 512; i += 128) {
        int r  = i >> 5;
        int c4 = (i & 31) << 2;
        float4 v = *(const float4*)(H + (size_t)(row0 + r) * DH + c4);
        *(float4*)&hs[r][c4] = v;
    }
    __syncthreads();

    const int wave = tid >> 5;
    const int lane = tid & 31;
    if (wave < 3) {                         // uniform per wave: EXEC stays all-1s
        const int n0    = wave * 16;
        const int m     = lane & 15;
        const int nn    = n0 + m;
        const int khalf = (lane >> 4) << 1;
        const bool inb  = (nn < DOUT);

        v8f c = {};
        for (int k0 = 0; k0 < DH; k0 += 4) {
            v2f a, b;
            a.x = hs[m][k0 + khalf];
            a.y = hs[m][k0 + khalf + 1];
            b.x = inb ? W[(size_t)(k0 + khalf)     * DOUT + nn] : 0.0f;
            b.y = inb ? W[(size_t)(k0 + khalf + 1) * DOUT + nn] : 0.0f;
            c = __builtin_amdgcn_wmma_f32_16x16x4_f32(false, a, false, b, (short)0, c, false, false);
        }
        if (inb) {
            const int rbase = row0 + ((lane >> 4) << 3);
#pragma unroll
            for (int i = 0; i < 8; ++i)
                Y[(size_t)(rbase + i) * DOUT + nn] = c[i];
        }
    }
}

// ---------------------------------------------------------------- edge aggregation D=128 (float4 chunks)
__global__ void agg128_kernel(const float* __restrict__ hlin,
                              const int* __restrict__ src, const int* __restrict__ dst,
                              const float* __restrict__ dinv, float* __restrict__ agg,
                              long long total) {
    long long id = (long long)blockIdx.x * blockDim.x + threadIdx.x;
    if (id >= total) return;                 // total = E * 32
    int e  = (int)(id >> 5);
    int c4 = (int)(id & 31) << 2;
    int s = src[e], d = dst[e];
    float coef = dinv[s] * dinv[d];
    float4 v = *(const float4*)(hlin + (size_t)s * DH + c4);
    float* o = agg + (size_t)d * DH + c4;
    atomicAdd(o + 0, v.x * coef);
    atomicAdd(o + 1, v.y * coef);
    atomicAdd(o + 2, v.z * coef);
    atomicAdd(o + 3, v.w * coef);
}

// ---------------------------------------------------------------- edge aggregation D=40 (float4 chunks, 10/edge)
__global__ void agg40_kernel(const float* __restrict__ hlin,
                             const int* __restrict__ src, const int* __restrict__ dst,
                             const float* __restrict__ dinv, float* __restrict__ agg,
                             long long total) {
    long long id = (long long)blockIdx.x * blockDim.x + threadIdx.x;
    if (id >= total) return;                 // total = E * 10
    int e  = (int)(id / 10);
    int c4 = (int)(id % 10) << 2;
    int s = src[e], d = dst[e];
    float coef = dinv[s] * dinv[d];
    float4 v = *(const float4*)(hlin + (size_t)s * DOUT + c4);
    float* o = agg + (size_t)d * DOUT + c4;
    atomicAdd(o + 0, v.x * coef);
    atomicAdd(o + 1, v.y * coef);
    atomicAdd(o + 2, v.z * coef);
    atomicAdd(o + 3, v.w * coef);
}

// ---------------------------------------------------------------- finalize: h = relu(agg + hlin*dinv^2 + b)
__global__ void finalize_kernel(const float* __restrict__ hlin, const float* __restrict__ b,
                                const float* __restrict__ dinv, float* __restrict__ agg,
                                int D, long long total) {
    long long id = (long long)blockIdx.x * blockDim.x + threadIdx.x;
    if (id >= total) return;
    int i = (int)(id / D);
    int c = (int)(id % D);
    float di = dinv[i];
    float v  = agg[id] + hlin[id] * di * di + b[c];
    agg[id] = fmaxf(v, 0.0f);
}

// ---------------------------------------------------------------- mean pool (sum via atomics)
__global__ void pool_kernel(const float* __restrict__ h, const int* __restrict__ batch,
                            float* __restrict__ pooled, float* __restrict__ cnt,
                            long long total) {
    long long id = (long long)blockIdx.x * blockDim.x + threadIdx.x;
    if (id >= total) return;                 // total = N * DOUT
    int i = (int)(id / DOUT);
    int c = (int)(id % DOUT);
    int g = batch[i];
    atomicAdd(&pooled[(size_t)g * DOUT + c], h[id]);
    if (c == 0) atomicAdd(&cnt[g], 1.0f);
}

// ---------------------------------------------------------------- log_softmax over 40 classes, one thread/graph
__global__ void lsm_kernel(const float* __restrict__ pooled, const float* __restrict__ cnt,
                           float* __restrict__ out) {
    int g = blockIdx.x * blockDim.x + threadIdx.x;
    if (g >= NGR) return;
    float inv = 1.0f / fmaxf(cnt[g], 1.0f);
    float mx = -1e30f;
    for (int c = 0; c < DOUT; ++c) mx = fmaxf(mx, pooled[(size_t)g * DOUT + c] * inv);
    float s = 0.0f;
    for (int c = 0; c < DOUT; ++c) s += expf(pooled[(size_t)g * DOUT + c] * inv - mx);
    float lse = mx + logf(s);
    for (int c = 0; c < DOUT; ++c) out[(size_t)g * DOUT + c] = pooled[(size_t)g * DOUT + c] * inv - lse;
}

// ================================================================ launcher
extern "C" void kernel_launch(void* const* d_in, const int* in_sizes, int n_in,
                              void* d_out, int out_size, void* d_ws, size_t ws_size,
                              hipStream_t stream) {
    const float* x     = (const float*)d_in[0];
    const int*   src   = (const int*)d_in[1];
    const int*   dst   = (const int*)d_in[2];
    const int*   batch = (const int*)d_in[3];
    const float* W1    = (const float*)d_in[4];
    const float* b1    = (const float*)d_in[5];
    const float* W2    = (const float*)d_in[6];
    const float* b2    = (const float*)d_in[7];
    float*       out   = (float*)d_out;

    const int N = in_sizes[3];      // batch has N entries
    const int E = in_sizes[1];

    // workspace carve-up (floats)
    float* ws     = (float*)d_ws;
    float* dinv   = ws;                                  // N       (deg -> dinv)
    float* h1lin  = dinv   + N;                          // N*128
    float* agg1   = h1lin  + (size_t)N * DH;             // N*128   (becomes h1)
    float* h2lin  = agg1   + (size_t)N * DH;             // N*40
    float* agg2   = h2lin  + (size_t)N * DOUT;           // N*40    (becomes h2)
    float* pooled = agg2   + (size_t)N * DOUT;           // G*40
    float* cnt    = pooled + (size_t)NGR * DOUT;         // G

    const int B = 256;

    // ---- init accumulators (must happen every call: graph replay)
    fill_kernel<<<(int)cdivll(N, B), B, 0, stream>>>(dinv, 1.0f, N);                       // self-loop
    fill_kernel<<<(int)cdivll((long long)N * DH, B), B, 0, stream>>>(agg1, 0.0f, (long long)N * DH);
    fill_kernel<<<(int)cdivll((long long)N * DOUT, B), B, 0, stream>>>(agg2, 0.0f, (long long)N * DOUT);
    fill_kernel<<<(int)cdivll((long long)NGR * DOUT + NGR, B), B, 0, stream>>>(pooled, 0.0f,
                                                                               (long long)NGR * DOUT + NGR);

    // ---- symmetric-norm coefficients
    deg_kernel<<<(int)cdivll(E, B), B, 0, stream>>>(dst, dinv, E);
    dinv_kernel<<<(int)cdivll(N, B), B, 0, stream>>>(dinv, N);

    // ---- layer 1
    gemm1_kernel<<<N / 16, 256, 0, stream>>>(x, W1, h1lin);
    {
        long long tot = (long long)E * 32;
        agg128_kernel<<<(int)cdivll(tot, B), B, 0, stream>>>(h1lin, src, dst, dinv, agg1, tot);
    }
    {
        long long tot = (long long)N * DH;
        finalize_kernel<<<(int)cdivll(tot, B), B, 0, stream>>>(h1lin, b1, dinv, agg1, DH, tot);
    }

    // ---- layer 2
    gemm2_kernel<<<N / 16, 128, 0, stream>>>(agg1, W2, h2lin);
    {
        long long tot = (long long)E * 10;
        agg40_kernel<<<(int)cdivll(tot, B), B, 0, stream>>>(h2lin, src, dst, dinv, agg2, tot);
    }
    {
        long long tot = (long long)N * DOUT;
        finalize_kernel<<<(int)cdivll(tot, B), B, 0, stream>>>(h2lin, b2, dinv, agg2, DOUT, tot);
    }

    // ---- pooling + log_softmax
    {
        long long tot = (long long)N * DOUT;
        pool_kernel<<<(int)cdivll(tot, B), B, 0, stream>>>(agg2, batch, pooled, cnt, tot);
    }
    lsm_kernel<<<(int)cdivll(NGR, B), B, 0, stream>>>(pooled, cnt, out);
}